// GCN_layer_8_56126632624286
// MI455X (gfx1250) — compile-verified
//
#include <hip/hip_runtime.h>

#define N_NODES 100000
#define N_EDGES 3200000

typedef float v2f __attribute__((ext_vector_type(2)));
typedef float v4f __attribute__((ext_vector_type(4)));
typedef float v8f __attribute__((ext_vector_type(8)));

// ---------------------------------------------------------------------------
// CSR build (once per launch, reused by all 8 layers)
// ---------------------------------------------------------------------------
__global__ void zero_int_kernel(int* __restrict__ p, int n) {
    int i = blockIdx.x * blockDim.x + threadIdx.x;
    if (i < n) p[i] = 0;
}

__global__ void count_deg_kernel(const int* __restrict__ col, int* __restrict__ deg, int ne) {
    int i = blockIdx.x * blockDim.x + threadIdx.x;
    if (i < ne) atomicAdd(&deg[col[i]], 1);
}

// Single-block sequential-chunk exclusive scan (runs once; ~100k ints).
__global__ void scan_exclusive_kernel(const int* __restrict__ deg,
                                      int* __restrict__ offs,
                                      int* __restrict__ cur, int n) {
    __shared__ int sh[1024];
    int carry = 0;
    for (int base = 0; base < n; base += 1024) {
        int i = base + (int)threadIdx.x;
        int v = (i < n) ? deg[i] : 0;
        sh[threadIdx.x] = v;
        __syncthreads();
        for (int off = 1; off < 1024; off <<= 1) {
            int t = (threadIdx.x >= (unsigned)off) ? sh[threadIdx.x - off] : 0;
            __syncthreads();
            sh[threadIdx.x] += t;
            __syncthreads();
        }
        int incl = sh[threadIdx.x];
        int excl = incl - v + carry;
        if (i < n) { offs[i] = excl; cur[i] = excl; }
        carry += sh[1023];
        __syncthreads();
    }
    if (threadIdx.x == 0) offs[n] = carry;
}

__global__ void scatter_edges_kernel(const int* __restrict__ col,
                                     int* __restrict__ cur,
                                     int* __restrict__ edst, int ne) {
    int i = blockIdx.x * blockDim.x + threadIdx.x;
    if (i < ne) {
        int c = col[i];
        int p = atomicAdd(&cur[c], 1);
        edst[p] = i;
    }
}

// ---------------------------------------------------------------------------
// GEMM: S = relu?(A) @ W   using V_WMMA_F32_16X16X4_F32
// Wave computes a 16x64 strip (4 accumulators). 8 waves/block -> 128 rows.
// Software-pipelined by one K-step: iteration i+1's fragment loads are issued
// before iteration i's WMMAs consume their fragments, so the loadcnt wait in
// front of the WMMA group covers a full iteration of load latency.
// All streams use uniform-base + per-lane unsigned element offsets so the
// backend emits saddr+voffset `scale_offset` loads with stable address regs.
// ---------------------------------------------------------------------------
template <bool RELU>
__global__ __launch_bounds__(256)
void gemm_wmma_f32_kernel(const float* __restrict__ A,
                          const float* __restrict__ W,
                          float* __restrict__ S,
                          int nrows, int K, int dout) {
    const int wave = threadIdx.x >> 5;
    const int lane = threadIdx.x & 31;
    const int row0 = (blockIdx.x * 8 + wave) * 16;
    const int col0 = blockIdx.y * 64;
    if (row0 >= nrows) return;   // wave-uniform exit

    const int m    = lane & 15;        // A: row within tile / B,C,D: col within tile
    const int half = lane >> 4;        // 0: lanes 0-15, 1: lanes 16-31
    const int koff = half * 2;         // A/B fragment K-pair selector

    const int r  = row0 + m;
    const int rc = (r < nrows) ? r : (nrows - 1);   // clamped row

    // A stream as v2f elements: offset = (rc*K + koff)/2 = rc*(K/2) + half.
    const v2f* __restrict__ A2 = (const v2f*)A;
    unsigned aoff = (unsigned)rc * (unsigned)(K >> 1) + (unsigned)half;

    const unsigned kb = (unsigned)koff * (unsigned)dout;
    unsigned bo0a, bo0b, bo1a, bo1b, bo2a, bo2b, bo3a, bo3b;
    {
        int n0 = col0 + 0 * 16 + m; n0 = (n0 < dout) ? n0 : (dout - 1);
        int n1 = col0 + 1 * 16 + m; n1 = (n1 < dout) ? n1 : (dout - 1);
        int n2 = col0 + 2 * 16 + m; n2 = (n2 < dout) ? n2 : (dout - 1);
        int n3 = col0 + 3 * 16 + m; n3 = (n3 < dout) ? n3 : (dout - 1);
        bo0a = kb + (unsigned)n0; bo0b = bo0a + (unsigned)dout;
        bo1a = kb + (unsigned)n1; bo1b = bo1a + (unsigned)dout;
        bo2a = kb + (unsigned)n2; bo2b = bo2a + (unsigned)dout;
        bo3a = kb + (unsigned)n3; bo3b = bo3a + (unsigned)dout;
    }
    const unsigned bstep = 4u * (unsigned)dout;

    v8f c0 = {}, c1 = {}, c2 = {}, c3 = {};

    // ---- pipeline stage 0: preload first fragments ----
    v2f a = A2[aoff];
    if (RELU) { a.x = fmaxf(a.x, 0.0f); a.y = fmaxf(a.y, 0.0f); }
    v2f b0, b1, b2, b3;
    b0.x = W[bo0a]; b0.y = W[bo0b];
    b1.x = W[bo1a]; b1.y = W[bo1b];
    b2.x = W[bo2a]; b2.y = W[bo2b];
    b3.x = W[bo3a]; b3.y = W[bo3b];
    aoff += 2u;
    bo0a += bstep; bo0b += bstep; bo1a += bstep; bo1b += bstep;
    bo2a += bstep; bo2b += bstep; bo3a += bstep; bo3b += bstep;

    // ---- steady state: load k0+4 fragments, then consume k0 fragments ----
#pragma unroll 2
    for (int k0 = 0; k0 < K - 4; k0 += 4) {
        v2f an = A2[aoff];
        if (RELU) { an.x = fmaxf(an.x, 0.0f); an.y = fmaxf(an.y, 0.0f); }
        v2f bn0, bn1, bn2, bn3;
        bn0.x = W[bo0a]; bn0.y = W[bo0b];
        bn1.x = W[bo1a]; bn1.y = W[bo1b];
        bn2.x = W[bo2a]; bn2.y = W[bo2b];
        bn3.x = W[bo3a]; bn3.y = W[bo3b];

        c0 = __builtin_amdgcn_wmma_f32_16x16x4_f32(false, a, false, b0, (short)0, c0, false, false);
        c1 = __builtin_amdgcn_wmma_f32_16x16x4_f32(false, a, false, b1, (short)0, c1, false, false);
        c2 = __builtin_amdgcn_wmma_f32_16x16x4_f32(false, a, false, b2, (short)0, c2, false, false);
        c3 = __builtin_amdgcn_wmma_f32_16x16x4_f32(false, a, false, b3, (short)0, c3, false, false);

        a = an; b0 = bn0; b1 = bn1; b2 = bn2; b3 = bn3;
        aoff += 2u;
        bo0a += bstep; bo0b += bstep; bo1a += bstep; bo1b += bstep;
        bo2a += bstep; bo2b += bstep; bo3a += bstep; bo3b += bstep;
    }

    // ---- epilogue: consume final fragments ----
    c0 = __builtin_amdgcn_wmma_f32_16x16x4_f32(false, a, false, b0, (short)0, c0, false, false);
    c1 = __builtin_amdgcn_wmma_f32_16x16x4_f32(false, a, false, b1, (short)0, c1, false, false);
    c2 = __builtin_amdgcn_wmma_f32_16x16x4_f32(false, a, false, b2, (short)0, c2, false, false);
    c3 = __builtin_amdgcn_wmma_f32_16x16x4_f32(false, a, false, b3, (short)0, c3, false, false);

#pragma unroll
    for (int t = 0; t < 4; ++t) {
        const int n = col0 + t * 16 + m;
        if (n >= dout) continue;
        const v8f c = (t == 0) ? c0 : (t == 1) ? c1 : (t == 2) ? c2 : c3;
#pragma unroll
        for (int i = 0; i < 8; ++i) {
            const int rr = row0 + i + 8 * half;   // C/D layout: VGPR i -> M = i + 8*half
            if (rr < nrows) S[(size_t)rr * dout + n] = c[i];
        }
    }
}

// ---------------------------------------------------------------------------
// CSR gather-aggregate: out[n,f] = b[f] + sum_{e: col[e]==n} w[e] * S[row[e], f]
// One block per destination node; thread owns 4 features (b128 gathers).
// Edge metadata loads are block-uniform -> scalar loads; 2-edge unroll keeps
// two independent b128 gathers in flight.
// ---------------------------------------------------------------------------
__global__ void csr_gather4_kernel(const float* __restrict__ S,
                                   const int* __restrict__ offs,
                                   const int* __restrict__ edst,
                                   const int* __restrict__ erow,
                                   const float* __restrict__ ew,
                                   const float* __restrict__ bias,
                                   float* __restrict__ out, int dout) {
    const int n = blockIdx.x;
    const int f = (int)threadIdx.x * 4;
    if (f >= dout) return;                 // no barriers in this kernel: legal
    v4f acc = *(const v4f*)(bias + f);
    const int beg = offs[n];
    const int end = offs[n + 1];
    int j = beg;
    for (; j + 1 < end; j += 2) {
        const int   e0 = edst[j];
        const int   e1 = edst[j + 1];
        const int   r0 = erow[e0];
        const int   r1 = erow[e1];
        const float w0 = ew[e0];
        const float w1 = ew[e1];
        const v4f   s0 = *(const v4f*)(S + (size_t)r0 * dout + f);
        const v4f   s1 = *(const v4f*)(S + (size_t)r1 * dout + f);
        acc += s0 * w0;
        acc += s1 * w1;
    }
    if (j < end) {
        const int   e = edst[j];
        const int   r = erow[e];
        const float w = ew[e];
        const v4f   s = *(const v4f*)(S + (size_t)r * dout + f);
        acc += s * w;
    }
    *(v4f*)(out + (size_t)n * dout + f) = acc;
}

// ---------------------------------------------------------------------------
// Launch
// ---------------------------------------------------------------------------
extern "C" void kernel_launch(void* const* d_in, const int* in_sizes, int n_in,
                              void* d_out, int out_size, void* d_ws, size_t ws_size,
                              hipStream_t stream) {
    const float* x    = (const float*)d_in[0];
    const int*   erow = (const int*)d_in[1];
    const int*   ecol = (const int*)d_in[2];
    const float* ew   = (const float*)d_in[3];
    const float* Wt[8];
    const float* bs[8];
    for (int i = 0; i < 8; ++i) {
        Wt[i] = (const float*)d_in[4 + 2 * i];
        bs[i] = (const float*)d_in[5 + 2 * i];
    }

    // Workspace carve-up (256B aligned)
    char*  ws  = (char*)d_ws;
    size_t off = 0;
    auto carve = [&](size_t bytes) -> void* {
        void* p = ws + off;
        off += (bytes + 255) & ~(size_t)255;
        return p;
    };
    float* bufS = (float*)carve((size_t)N_NODES * 256 * sizeof(float));
    float* bufX = (float*)carve((size_t)N_NODES * 256 * sizeof(float));
    int*   deg  = (int*)carve((size_t)N_NODES * sizeof(int));
    int*   offs = (int*)carve((size_t)(N_NODES + 1) * sizeof(int));
    int*   cur  = (int*)carve((size_t)N_NODES * sizeof(int));
    int*   edst = (int*)carve((size_t)N_EDGES * sizeof(int));

    // CSR build (once; reused by all layers)
    zero_int_kernel<<<(N_NODES + 255) / 256, 256, 0, stream>>>(deg, N_NODES);
    count_deg_kernel<<<(N_EDGES + 255) / 256, 256, 0, stream>>>(ecol, deg, N_EDGES);
    scan_exclusive_kernel<<<1, 1024, 0, stream>>>(deg, offs, cur, N_NODES);
    scatter_edges_kernel<<<(N_EDGES + 255) / 256, 256, 0, stream>>>(ecol, cur, edst, N_EDGES);

    const int dims_in[8]  = {512, 256, 256, 256, 256, 256, 256, 256};
    const int dims_out[8] = {256, 256, 256, 256, 256, 256, 256, 40};

    const float* h = x;
    for (int L = 0; L < 8; ++L) {
        const int K    = dims_in[L];
        const int dout = dims_out[L];
        dim3 gg((N_NODES + 127) / 128, (dout + 63) / 64);
        if (L == 0)
            gemm_wmma_f32_kernel<false><<<gg, 256, 0, stream>>>(h, Wt[L], bufS, N_NODES, K, dout);
        else
            gemm_wmma_f32_kernel<true><<<gg, 256, 0, stream>>>(h, Wt[L], bufS, N_NODES, K, dout);

        float* outp = (L == 7) ? (float*)d_out : bufX;
        const int nthreads4 = (dout + 3) / 4;                  // threads owning 4 feats
        const int blk = ((nthreads4 + 31) / 32) * 32;          // round to wave
        csr_gather4_kernel<<<N_NODES, blk, 0, stream>>>(bufS, offs, edst, erow, ew, bs[L], outp, dout);
        h = bufX;
    }
}